// GCN_6554120094214
// MI455X (gfx1250) — compile-verified
//
#include <hip/hip_runtime.h>

typedef float v2f __attribute__((ext_vector_type(2)));
typedef float v8f __attribute__((ext_vector_type(8)));

#define HDIM 64
#define EPSV 1e-5f

static inline int cdiv(long long a, long long b) { return (int)((a + b - 1) / b); }

// ---------------- utility ----------------
__global__ void k_zero(float* __restrict__ p, int n) {
    int i = blockIdx.x * blockDim.x + threadIdx.x;
    if (i < n) p[i] = 0.0f;
}

__global__ void k_deg_init(float* __restrict__ deg, int n) {
    int i = blockIdx.x * blockDim.x + threadIdx.x;
    if (i < n) deg[i] = 1.0f;  // self-loop contributes 1 to every node's degree
}

__global__ void k_deg_accum(const int* __restrict__ dst, float* __restrict__ deg, int E) {
    int e = blockIdx.x * blockDim.x + threadIdx.x;
    if (e < E) atomicAdd(&deg[dst[e]], 1.0f);
}

__global__ void k_dinv(float* __restrict__ deg, int n) {
    int i = blockIdx.x * blockDim.x + threadIdx.x;
    if (i < n) deg[i] = rsqrtf(deg[i]);  // deg >= 1 always (self-loops)
}

// ---------------- FP32 WMMA GEMM: out[n,o] = sum_k A[n,k] * W[o,k] ----------------
// One wave32 per 16x16 output tile. K loop: 16 x V_WMMA_F32_16X16X4_F32.
// A 16x4 f32 layout: lanes 0-15 -> M=lane, K=0(v0),1(v1); lanes 16-31 -> M=lane-16, K=2,3.
// B 4x16 f32 layout mirrors: lane n holds B[kh..kh+1, n] = W[n, kh..kh+1] (contiguous f32x2).
// D 16x16 f32: VGPR r -> row r (lanes 0-15) / row r+8 (lanes 16-31), col = lane%16.
__global__ void k_gemm_wmma(const float* __restrict__ A, const float* __restrict__ W,
                            float* __restrict__ out, int rowTiles) {
    int wid  = (blockIdx.x * blockDim.x + threadIdx.x) >> 5;
    int lane = threadIdx.x & 31;
    int tr = wid >> 2;   // row tile (0 .. rowTiles-1)
    int tc = wid & 3;    // col tile (0 .. 3) — HDIM/16
    if (tr >= rowTiles) return;  // wave-uniform: EXEC stays all-1s for WMMA
    int m  = lane & 15;
    int kh = (lane >> 4) << 1;   // 0 or 2
    const float* Ar = A + (size_t)(tr * 16 + m) * HDIM;
    const float* Wr = W + (size_t)(tc * 16 + m) * HDIM;  // W[o,k] row-major, o = col of tile
    v8f c = {0.f, 0.f, 0.f, 0.f, 0.f, 0.f, 0.f, 0.f};
#pragma unroll
    for (int k0 = 0; k0 < HDIM; k0 += 4) {
        v2f a = *(const v2f*)(Ar + k0 + kh);
        v2f b = *(const v2f*)(Wr + k0 + kh);
        c = __builtin_amdgcn_wmma_f32_16x16x4_f32(false, a, false, b, (short)0, c, false, false);
    }
    int rowBase = tr * 16 + ((lane >> 4) << 3);
    int col     = tc * 16 + m;
#pragma unroll
    for (int r = 0; r < 8; ++r)
        out[(size_t)(rowBase + r) * HDIM + col] = c[r];
}

// ---------------- self-loop init: agg[n,:] = hW[n,:] * dinv[n]^2 (full overwrite) ------
__global__ void k_self_init(const float* __restrict__ hW, const float* __restrict__ dinv,
                            float* __restrict__ agg, int N) {
    int t = blockIdx.x * blockDim.x + threadIdx.x;
    if (t >= N * 16) return;
    int n = t >> 4, q = t & 15;
    float di  = dinv[n];
    float nrm = di * di;
    float4 v  = ((const float4*)(hW + (size_t)n * HDIM))[q];
    ((float4*)(agg + (size_t)n * HDIM))[q] =
        make_float4(v.x * nrm, v.y * nrm, v.z * nrm, v.w * nrm);
}

// ---------------- edge scatter: agg[dst,:] += hW[src,:] * dinv[src]*dinv[dst] ----------
__global__ void k_scatter(const float* __restrict__ hW, const float* __restrict__ dinv,
                          const int* __restrict__ src, const int* __restrict__ dst,
                          float* __restrict__ agg, int E) {
    int t = blockIdx.x * blockDim.x + threadIdx.x;
    if (t >= E * 16) return;
    int e = t >> 4, q = t & 15;
    int s = src[e], d = dst[e];
    float nrm = dinv[s] * dinv[d];
    float4 v = ((const float4*)(hW + (size_t)s * HDIM))[q];
    float* a = agg + (size_t)d * HDIM + q * 4;
    atomicAdd(a + 0, v.x * nrm);
    atomicAdd(a + 1, v.y * nrm);
    atomicAdd(a + 2, v.z * nrm);
    atomicAdd(a + 3, v.w * nrm);
}

// ---------------- fused bias + ReLU (in place) + per-feature sum/sumsq ----------------
__global__ void k_bias_relu_stats(float* __restrict__ agg, const float* __restrict__ bias,
                                  float* __restrict__ stats /*[128]*/, int N) {
    __shared__ float sS[256];
    __shared__ float sQ[256];
    int f  = threadIdx.x & 63;
    int rg = threadIdx.x >> 6;  // 4 row-groups per block
    float b = bias[f];
    float s = 0.f, q = 0.f;
    for (int row = blockIdx.x * 4 + rg; row < N; row += gridDim.x * 4) {
        size_t idx = (size_t)row * HDIM + f;
        float v = agg[idx] + b;
        v = fmaxf(v, 0.f);
        agg[idx] = v;
        s += v;
        q += v * v;
    }
    sS[threadIdx.x] = s;
    sQ[threadIdx.x] = q;
    __syncthreads();
    if (rg == 0) {
        s = sS[f] + sS[64 + f] + sS[128 + f] + sS[192 + f];
        q = sQ[f] + sQ[64 + f] + sQ[128 + f] + sQ[192 + f];
        atomicAdd(&stats[f], s);
        atomicAdd(&stats[64 + f], q);
    }
}

// ---------------- fused BatchNorm apply + per-graph add-pool ----------------
__global__ void k_bn_pool(const float* __restrict__ hin, const float* __restrict__ stats,
                          const float* __restrict__ g, const float* __restrict__ beta,
                          const int* __restrict__ batch, float* __restrict__ hout,
                          float* __restrict__ pooled, int poolOff, int N) {
    int t = blockIdx.x * blockDim.x + threadIdx.x;
    if (t >= N * HDIM) return;
    int n = t >> 6, f = t & 63;
    float invN = 1.0f / (float)N;
    float mean = stats[f] * invN;
    float var  = stats[64 + f] * invN - mean * mean;  // biased batch variance
    float v = g[f] * (hin[t] - mean) * rsqrtf(var + EPSV) + beta[f];
    hout[t] = v;
    atomicAdd(&pooled[(size_t)batch[n] * 192 + poolOff + f], v);
}

extern "C" void kernel_launch(void* const* d_in, const int* in_sizes, int n_in,
                              void* d_out, int out_size, void* d_ws, size_t ws_size,
                              hipStream_t stream) {
    const float* x    = (const float*)d_in[0];
    const int*   eidx = (const int*)d_in[1];
    const int*   batch = (const int*)d_in[2];
    const float* W[3]  = {(const float*)d_in[3],  (const float*)d_in[7],  (const float*)d_in[11]};
    const float* bi[3] = {(const float*)d_in[4],  (const float*)d_in[8],  (const float*)d_in[12]};
    const float* ga[3] = {(const float*)d_in[5],  (const float*)d_in[9],  (const float*)d_in[13]};
    const float* be[3] = {(const float*)d_in[6],  (const float*)d_in[10], (const float*)d_in[14]};

    const int N = in_sizes[0] / HDIM;       // 100000
    const int E = in_sizes[1] / 2;          // 1000000
    const int G = (out_size - N * HDIM) / (3 * HDIM);  // 512

    const int* src = eidx;          // edge_index[0]
    const int* dst = eidx + E;      // edge_index[1]

    // workspace layout (floats); all offsets 16B-aligned
    float* ws    = (float*)d_ws;
    float* dinv  = ws;                       // N
    float* tmp   = dinv + N;                 // N*64  (GEMM output hW)
    float* agg   = tmp  + (size_t)N * HDIM;  // N*64  (aggregation, then relu'd h)
    float* bufH  = agg  + (size_t)N * HDIM;  // N*64  (BN output / next layer input)
    float* stats = bufH + (size_t)N * HDIM;  // 128   (sum | sumsq)

    float* pooled = (float*)d_out;             // [G, 192]
    float* hfinal = (float*)d_out + (size_t)G * 192;  // [N, 64]

    const int TB = 256;

    // degree -> dinv (in place)
    k_deg_init <<<cdiv(N, TB), TB, 0, stream>>>(dinv, N);
    k_deg_accum<<<cdiv(E, TB), TB, 0, stream>>>(dst, dinv, E);
    k_dinv     <<<cdiv(N, TB), TB, 0, stream>>>(dinv, N);

    // zero pooled output region
    k_zero<<<cdiv((long long)G * 192, TB), TB, 0, stream>>>(pooled, G * 192);

    const int rowTiles   = N / 16;                 // 6250 (exact)
    const int gemmWaves  = rowTiles * (HDIM / 16); // 25000
    const int gemmBlocks = cdiv((long long)gemmWaves * 32, TB);

    const float* gin = x;
    for (int l = 0; l < 3; ++l) {
        float* hout = (l == 2) ? hfinal : bufH;

        k_gemm_wmma<<<gemmBlocks, TB, 0, stream>>>(gin, W[l], tmp, rowTiles);
        k_self_init<<<cdiv((long long)N * 16, TB), TB, 0, stream>>>(tmp, dinv, agg, N);
        k_scatter  <<<cdiv((long long)E * 16, TB), TB, 0, stream>>>(tmp, dinv, src, dst, agg, E);

        k_zero<<<1, 128, 0, stream>>>(stats, 128);
        k_bias_relu_stats<<<1024, TB, 0, stream>>>(agg, bi[l], stats, N);

        k_bn_pool<<<cdiv((long long)N * HDIM, TB), TB, 0, stream>>>(
            agg, stats, ga[l], be[l], batch, hout, pooled, l * HDIM, N);

        gin = hout;
    }
}